// LSTMLayer_78615081386411
// MI455X (gfx1250) — compile-verified
//
#include <hip/hip_runtime.h>
#include <hip/hip_bf16.h>

#define BB 32
#define TT 256
#define DD 1024
#define HH 1024
#define GG 4096          // 4*H
#define MM 8192          // B*T

typedef __attribute__((ext_vector_type(2))) float v2f;
typedef __attribute__((ext_vector_type(8))) float v8f;

__device__ __forceinline__ float sigf(float x) {
    return 1.0f / (1.0f + __expf(-x));
}

__device__ __forceinline__ v8f wmma_f32(v2f a, v2f b, v8f c) {
    return __builtin_amdgcn_wmma_f32_16x16x4_f32(false, a, false, b, (short)0, c,
                                                 false, false);
}

// ---------------------------------------------------------------- lengths[b]
__global__ void k_lengths(const int* __restrict__ mask, int* __restrict__ len) {
    __shared__ int s[256];
    const int b = blockIdx.x;
    const int t = threadIdx.x;
    s[t] = mask[b * TT + t];
    __syncthreads();
    for (int o = 128; o > 0; o >>= 1) {
        if (t < o) s[t] += s[t + o];
        __syncthreads();
    }
    if (t == 0) len[b] = s[0];
}

// ---------------------------------------------------------------- zero init
__global__ void k_zero(float* __restrict__ p, int n) {
    int i = blockIdx.x * blockDim.x + threadIdx.x;
    if (i < n) p[i] = 0.0f;
}

// --------------------------------------------- xg = x(_rev) @ Wi + b  (both dirs)
// grid: (M/64, G/256, 2), block 256 (8 waves). Wave tile = 32x64 (2x4 WMMA tiles)
// -> A-fragment reuse x4, 12 loaded dwords per 8 WMMAs.
__global__ void k_xg(const float* __restrict__ x,
                     const float* __restrict__ Wi_f, const float* __restrict__ Wi_b,
                     const float* __restrict__ bf,  const float* __restrict__ bbk,
                     const int* __restrict__ len, float* __restrict__ xg) {
    const int dir = blockIdx.z;
    const float* __restrict__ Wi   = dir ? Wi_b : Wi_f;
    const float* __restrict__ bias = dir ? bbk  : bf;

    const int tid  = threadIdx.x;
    const int lane = tid & 31;
    const int wave = tid >> 5;
    const int l16  = lane & 15;
    const int lh   = lane >> 4;
    const int wm   = wave >> 2;     // 0..1
    const int wn   = wave & 3;      // 0..3
    const int m0   = blockIdx.x * 64 + wm * 32;
    const int n0   = blockIdx.y * 256 + wn * 64;

    // Per-M-tile source row base (time-flipped for backward direction).
    const float* arow[2];
#pragma unroll
    for (int mt = 0; mt < 2; ++mt) {
        int row = m0 + mt * 16 + l16;       // output row = b*T + t
        int b = row >> 8;                   // T = 256
        int tq = row & 255;
        if (dir) tq = (TT - 1 - tq + len[b]) % TT;
        arow[mt] = x + ((size_t)b * TT + tq) * DD;
    }

    v8f acc[2][4];
#pragma unroll
    for (int mt = 0; mt < 2; ++mt)
#pragma unroll
        for (int nt = 0; nt < 4; ++nt) {
            float bv = bias[n0 + nt * 16 + l16];
#pragma unroll
            for (int v = 0; v < 8; ++v) acc[mt][nt][v] = bv;
        }

    for (int k = 0; k < DD; k += 4) {
        v2f a[2], bw[4];
#pragma unroll
        for (int mt = 0; mt < 2; ++mt)
            a[mt] = *(const v2f*)(arow[mt] + k + lh * 2);
#pragma unroll
        for (int nt = 0; nt < 4; ++nt) {
            const float* p = Wi + (size_t)(k + lh * 2) * GG + n0 + nt * 16 + l16;
            bw[nt][0] = p[0];
            bw[nt][1] = p[GG];
        }
#pragma unroll
        for (int mt = 0; mt < 2; ++mt)
#pragma unroll
            for (int nt = 0; nt < 4; ++nt)
                acc[mt][nt] = wmma_f32(a[mt], bw[nt], acc[mt][nt]);
    }

    float* outb = xg + (size_t)dir * MM * GG;
#pragma unroll
    for (int mt = 0; mt < 2; ++mt)
#pragma unroll
        for (int nt = 0; nt < 4; ++nt) {
            int col = n0 + nt * 16 + l16;
#pragma unroll
            for (int v = 0; v < 8; ++v) {
                int row = m0 + mt * 16 + v + 8 * lh;
                outb[(size_t)row * GG + col] = acc[mt][nt][v];
            }
        }
}

// -------------------------------------------------- one LSTM time step, both dirs
// grid: 128 blocks x 256 threads = 1024 waves.
// 4 waves cooperate on each (dir, batch-tile, h-col-tile): 4-way K-split with
// LDS reduction, then each wave finalizes 1/4 of the tile's cell updates.
__global__ void k_step(int t, int use_xg,
                       const float* __restrict__ x,
                       const float* __restrict__ Wi_f, const float* __restrict__ Wi_b,
                       const float* __restrict__ Wh_f, const float* __restrict__ Wh_b,
                       const float* __restrict__ bf,   const float* __restrict__ bbk,
                       const float* __restrict__ xg,
                       const float* __restrict__ hprev, float* __restrict__ hnext,
                       float* __restrict__ c, const int* __restrict__ len,
                       float* __restrict__ out) {
    // red[tileLocal][kslice][gate][v*32+lane]  (conflict-free on both sides)
    __shared__ float red[2 * 4 * 4 * 256];

    const int tid   = threadIdx.x;
    const int lane  = tid & 31;
    const int l16   = lane & 15;
    const int lh    = lane >> 4;
    const int wave  = tid >> 5;              // 0..7
    const int tileL = wave >> 2;             // 0..1
    const int ks    = wave & 3;              // K slice 0..3
    const int tile  = blockIdx.x * 2 + tileL;  // 0..255
    const int dir   = tile >> 7;
    const int wd    = tile & 127;
    const int mt    = wd >> 6;               // batch tile 0..1
    const int nt    = wd & 63;               // h-col tile 0..63
    const int m0    = mt * 16;
    const int n0    = nt * 16;

    const float* __restrict__ Wh   = dir ? Wh_b : Wh_f;
    const float* __restrict__ Wi   = dir ? Wi_b : Wi_f;
    const float* __restrict__ bias = dir ? bbk  : bf;

    v8f acc[4];
#pragma unroll
    for (int g = 0; g < 4; ++g)
#pragma unroll
        for (int v = 0; v < 8; ++v) acc[g][v] = 0.0f;

    if (!use_xg) {
        // fused x_t @ Wi, this wave's K slice of D
        int brow = m0 + l16;
        int tq = t;
        if (dir) tq = (TT - 1 - t + len[brow]) % TT;
        const float* arow = x + ((size_t)brow * TT + tq) * DD;
        const int kb = ks * (DD / 4);
        for (int k = kb; k < kb + DD / 4; k += 4) {
            v2f a = *(const v2f*)(arow + k + lh * 2);
#pragma unroll
            for (int g = 0; g < 4; ++g) {
                const float* p = Wi + (size_t)(k + lh * 2) * GG + g * HH + n0 + l16;
                v2f bw;
                bw[0] = p[0];
                bw[1] = p[GG];
                acc[g] = wmma_f32(a, bw, acc[g]);
            }
        }
    }

    // recurrent h_{t-1} @ Wh, this wave's K slice of H
    {
        const float* arow = hprev + ((size_t)dir * BB + (m0 + l16)) * HH;
        const int kb = ks * (HH / 4);
        for (int k = kb; k < kb + HH / 4; k += 4) {
            v2f a = *(const v2f*)(arow + k + lh * 2);
#pragma unroll
            for (int g = 0; g < 4; ++g) {
                const float* p = Wh + (size_t)(k + lh * 2) * GG + g * HH + n0 + l16;
                v2f bw;
                bw[0] = p[0];
                bw[1] = p[GG];
                acc[g] = wmma_f32(a, bw, acc[g]);
            }
        }
    }

    // write partials to LDS
#pragma unroll
    for (int g = 0; g < 4; ++g) {
        float* r = &red[((tileL * 4 + ks) * 4 + g) * 256];
#pragma unroll
        for (int v = 0; v < 8; ++v) r[v * 32 + lane] = acc[g][v];
    }
    __syncthreads();

    // this wave finalizes v in {2*ks, 2*ks+1} of its tile
#pragma unroll
    for (int v2 = 0; v2 < 2; ++v2) {
        const int v = 2 * ks + v2;
        float z[4] = {0.0f, 0.0f, 0.0f, 0.0f};
#pragma unroll
        for (int k2 = 0; k2 < 4; ++k2)
#pragma unroll
            for (int g = 0; g < 4; ++g)
                z[g] += red[((tileL * 4 + k2) * 4 + g) * 256 + v * 32 + lane];

        const int b = m0 + v + 8 * lh;
        const int n = n0 + l16;
        if (use_xg) {
#pragma unroll
            for (int g = 0; g < 4; ++g)
                z[g] += xg[((size_t)dir * MM + (size_t)b * TT + t) * GG +
                           g * HH + n];
        } else {
#pragma unroll
            for (int g = 0; g < 4; ++g) z[g] += bias[g * HH + n];
        }

        size_t sidx = (size_t)dir * BB * HH + (size_t)b * HH + n;
        float iv = sigf(z[0]);
        float fv = sigf(z[1]);
        float gv = tanhf(z[2]);
        float ov = sigf(z[3]);
        float cn = fv * c[sidx] + iv * gv;
        c[sidx] = cn;
        float hv = ov * tanhf(cn);
        hnext[sidx] = hv;
        int td = dir ? ((TT - 1 - t + len[b]) % TT) : t;
        out[((size_t)b * TT + td) * (2 * HH) + dir * HH + n] = hv;
    }
}

extern "C" void kernel_launch(void* const* d_in, const int* in_sizes, int n_in,
                              void* d_out, int out_size, void* d_ws, size_t ws_size,
                              hipStream_t stream) {
    const float* x    = (const float*)d_in[0];
    const int*   mask = (const int*)d_in[1];
    const float* Wi_f = (const float*)d_in[2];
    const float* Wh_f = (const float*)d_in[3];
    const float* b_f  = (const float*)d_in[4];
    const float* Wi_b = (const float*)d_in[5];
    const float* Wh_b = (const float*)d_in[6];
    const float* b_b  = (const float*)d_in[7];
    float* out = (float*)d_out;

    char* ws = (char*)d_ws;
    size_t off = 0;
    int* len = (int*)(ws + off);      off += 256;
    float* c  = (float*)(ws + off);   off += 2ull * BB * HH * 4;
    float* h0 = (float*)(ws + off);   off += 2ull * BB * HH * 4;
    float* h1 = (float*)(ws + off);   off += 2ull * BB * HH * 4;
    size_t xg_bytes = 2ull * MM * GG * 4;   // 256 MB
    float* xg = (float*)(ws + off);
    const int use_xg = (ws_size >= off + xg_bytes) ? 1 : 0;

    k_lengths<<<BB, 256, 0, stream>>>(mask, len);
    // c and h0 are contiguous: zero both (state init h=0, c=0)
    int zn = 4 * BB * HH;
    k_zero<<<(zn + 255) / 256, 256, 0, stream>>>(c, zn);

    if (use_xg)
        k_xg<<<dim3(MM / 64, GG / 256, 2), 256, 0, stream>>>(x, Wi_f, Wi_b, b_f,
                                                             b_b, len, xg);

    for (int t = 0; t < TT; ++t) {
        const float* hp = (t & 1) ? h1 : h0;
        float*       hn = (t & 1) ? h0 : h1;
        k_step<<<128, 256, 0, stream>>>(t, use_xg, x, Wi_f, Wi_b, Wh_f, Wh_b,
                                        b_f, b_b, xg, hp, hn, c, len, out);
    }
}